// QLSTMGen_65481071405044
// MI455X (gfx1250) — compile-verified
//
#include <hip/hip_runtime.h>

typedef float v2f __attribute__((ext_vector_type(2)));
typedef float v8f __attribute__((ext_vector_type(8)));
typedef float v4f __attribute__((ext_vector_type(4)));

#define SEQ_LEN   128
#define BATCH     512
#define INPUT_DIM 64
#define NQ        8
#define NGATE     4
#define NCOL      32          // 4 gates * 8 wires
#define FAN_IN    72
#define MROWS     (SEQ_LEN * BATCH)   // 65536

// ---------------------------------------------------------------------------
// Pack W_{f,i,u,o} (8 x 72 each) into:
//   Wall[32][64]  : input-projection weights, n = gate*8 + wire, k-contiguous
//   Whid[32][8]   : hidden-projection weights
//   bsum[32]      : b_g + th_g fused bias
// ---------------------------------------------------------------------------
__global__ __launch_bounds__(256) void qlstm_pack(
    const float* __restrict__ Wf, const float* __restrict__ bf, const float* __restrict__ thf,
    const float* __restrict__ Wi, const float* __restrict__ bi, const float* __restrict__ thi,
    const float* __restrict__ Wu, const float* __restrict__ bu, const float* __restrict__ thu,
    const float* __restrict__ Wo, const float* __restrict__ bo, const float* __restrict__ tho,
    float* __restrict__ Wall, float* __restrict__ Whid, float* __restrict__ bsum)
{
    const int tid = threadIdx.x;
    for (int e = tid; e < NCOL * INPUT_DIM; e += 256) {
        const int n = e / INPUT_DIM, k = e % INPUT_DIM;
        const int g = n >> 3, w = n & 7;
        const float* W = (g == 0) ? Wf : (g == 1) ? Wi : (g == 2) ? Wu : Wo;
        Wall[e] = W[w * FAN_IN + k];
    }
    for (int e = tid; e < NCOL * NQ; e += 256) {
        const int n = e / NQ, j = e % NQ;
        const int g = n >> 3, w = n & 7;
        const float* W = (g == 0) ? Wf : (g == 1) ? Wi : (g == 2) ? Wu : Wo;
        Whid[e] = W[w * FAN_IN + INPUT_DIM + j];
    }
    if (tid < NCOL) {
        const int g = tid >> 3, w = tid & 7;
        const float* b  = (g == 0) ? bf  : (g == 1) ? bi  : (g == 2) ? bu  : bo;
        const float* th = (g == 0) ? thf : (g == 1) ? thi : (g == 2) ? thu : tho;
        bsum[tid] = b[w] + th[w];
    }
}

// ---------------------------------------------------------------------------
// Input projection: XP[m][n] = X[m][:] . Wall[n][:] + bsum[n]
// M = 65536, N = 32, K = 64.  One wave per 16-row M-tile, two N-tiles,
// 16 K-steps of V_WMMA_F32_16X16X4_F32 (f32 in / f32 accum, matches ref).
// ---------------------------------------------------------------------------
#if defined(__has_builtin)
#  if __has_builtin(__builtin_amdgcn_wmma_f32_16x16x4_f32)
#    define HAVE_WMMA_F32X4 1
#  endif
#endif

#ifdef HAVE_WMMA_F32X4
__global__ __launch_bounds__(256) void qlstm_xproj_wmma(
    const float* __restrict__ X, const float* __restrict__ Wall,
    const float* __restrict__ bsum, float* __restrict__ XP)
{
    const int wave  = (blockIdx.x * 256 + threadIdx.x) >> 5;   // 0..4095
    const int lane  = threadIdx.x & 31;
    const int lhalf = lane & 15;
    const int hi    = lane >> 4;           // 0: K={0,1}, 1: K={2,3}
    const int row0  = wave * 16;

    // A frag: 16x4 f32 tile of X. lane<16 -> (row0+lane, K 0..1); lane>=16 -> (.., K 2..3)
    const float* aptr  = X    + (size_t)(row0 + lhalf) * INPUT_DIM + 2 * hi;
    // B frags: 4x16 f32 tiles of Wall (K x N), N = col
    const float* bptr0 = Wall + (size_t)(lhalf)      * INPUT_DIM + 2 * hi;
    const float* bptr1 = Wall + (size_t)(16 + lhalf) * INPUT_DIM + 2 * hi;

    v8f acc0 = {0.f, 0.f, 0.f, 0.f, 0.f, 0.f, 0.f, 0.f};
    v8f acc1 = {0.f, 0.f, 0.f, 0.f, 0.f, 0.f, 0.f, 0.f};
#pragma unroll
    for (int kb = 0; kb < 16; ++kb) {
        const v2f a  = *(const v2f*)(aptr  + kb * 4);
        const v2f b0 = *(const v2f*)(bptr0 + kb * 4);
        const v2f b1 = *(const v2f*)(bptr1 + kb * 4);
        acc0 = __builtin_amdgcn_wmma_f32_16x16x4_f32(false, a, false, b0, (short)0, acc0, false, false);
        acc1 = __builtin_amdgcn_wmma_f32_16x16x4_f32(false, a, false, b1, (short)0, acc1, false, false);
    }
    const float bias0 = bsum[lhalf];
    const float bias1 = bsum[16 + lhalf];
    // C/D layout: VGPR r -> row (r + 8*hi), col = lhalf (+16 for tile 1)
#pragma unroll
    for (int r = 0; r < 8; ++r) {
        const size_t row = (size_t)(row0 + r + 8 * hi);
        XP[row * NCOL + lhalf]      = acc0[r] + bias0;
        XP[row * NCOL + 16 + lhalf] = acc1[r] + bias1;
    }
}
#else
__global__ __launch_bounds__(256) void qlstm_xproj_plain(
    const float* __restrict__ X, const float* __restrict__ Wall,
    const float* __restrict__ bsum, float* __restrict__ XP)
{
    const int idx = blockIdx.x * 256 + threadIdx.x;    // over MROWS*NCOL
    const int row = idx >> 5, col = idx & 31;
    const float* xr = X + (size_t)row * INPUT_DIM;
    const float* wr = Wall + (size_t)col * INPUT_DIM;
    float s = bsum[col];
#pragma unroll
    for (int k = 0; k < INPUT_DIM; ++k) s = fmaf(xr[k], wr[k], s);
    XP[idx] = s;
}
#endif

// ---------------------------------------------------------------------------
// Sequential LSTM recurrence, one thread per batch element.
// Quantum layer collapses to: q_w = prod_{v in S_w} cos(ang_v)  (see analysis).
// ---------------------------------------------------------------------------
__device__ __forceinline__ float sigmoid_f(float x) {
    return 1.f / (1.f + __expf(-x));
}
__device__ __forceinline__ float tanh_f(float x) {
    const float e = __expf(-2.f * x);
    return (1.f - e) / (1.f + e);
}

__global__ __launch_bounds__(256) void qlstm_recurrence(
    const float* __restrict__ XP, const float* __restrict__ Whid,
    float* __restrict__ out)
{
    __shared__ float wh[NCOL * NQ];
    for (int e = threadIdx.x; e < NCOL * NQ; e += 256) wh[e] = Whid[e];
    __syncthreads();

    const int b = blockIdx.x * 256 + threadIdx.x;   // 0..511

    float hx[NQ], cx[NQ];
#pragma unroll
    for (int w = 0; w < NQ; ++w) { hx[w] = 0.f; cx[w] = 0.f; }

    // Wire-subset masks from composing the depth-2 CNOT ring over GF(2).
    const int SM[NQ] = {0xAB, 0xFD, 0xFA, 0xF5, 0xEA, 0xD5, 0xAA, 0x55};

    for (int t = 0; t < SEQ_LEN; ++t) {
        const float* xp = XP + ((size_t)t * BATCH + b) * NCOL;

        float qc[NCOL];
#pragma unroll
        for (int n = 0; n < NCOL; ++n) {
            float s = xp[n];
#pragma unroll
            for (int j = 0; j < NQ; ++j) s = fmaf(hx[j], wh[n * NQ + j], s);
            qc[n] = __cosf(s);
        }

        float q[NGATE][NQ];
#pragma unroll
        for (int g = 0; g < NGATE; ++g) {
#pragma unroll
            for (int w = 0; w < NQ; ++w) {
                float p = 1.f;
#pragma unroll
                for (int v = 0; v < NQ; ++v)
                    if (SM[w] & (1 << v)) p *= qc[g * NQ + v];
                q[g][w] = p;
            }
        }

#pragma unroll
        for (int w = 0; w < NQ; ++w) {
            const float fg = sigmoid_f(q[0][w]);
            const float ig = sigmoid_f(q[1][w]);
            const float gg = tanh_f(q[2][w]);
            const float og = sigmoid_f(q[3][w]);
            const float c  = fg * cx[w] + ig * gg;
            cx[w] = c;
            hx[w] = og * tanh_f(c);
        }

        float* op = out + ((size_t)t * BATCH + b) * NQ;
#pragma unroll
        for (int w = 0; w < NQ; ++w) op[w] = hx[w];
    }

    // Tuple tail: hx then cx, each (BATCH, NQ)
    float* hp = out + (size_t)SEQ_LEN * BATCH * NQ + (size_t)b * NQ;
    float* cp = hp + (size_t)BATCH * NQ;
#pragma unroll
    for (int w = 0; w < NQ; ++w) { hp[w] = hx[w]; cp[w] = cx[w]; }
}

// ---------------------------------------------------------------------------
extern "C" void kernel_launch(void* const* d_in, const int* in_sizes, int n_in,
                              void* d_out, int out_size, void* d_ws, size_t ws_size,
                              hipStream_t stream)
{
    const float* X   = (const float*)d_in[0];
    const float* Wf  = (const float*)d_in[1];
    const float* bf  = (const float*)d_in[2];
    const float* thf = (const float*)d_in[3];
    const float* Wi  = (const float*)d_in[4];
    const float* bi  = (const float*)d_in[5];
    const float* thi = (const float*)d_in[6];
    const float* Wu  = (const float*)d_in[7];
    const float* bu  = (const float*)d_in[8];
    const float* thu = (const float*)d_in[9];
    const float* Wo  = (const float*)d_in[10];
    const float* bo  = (const float*)d_in[11];
    const float* tho = (const float*)d_in[12];

    float* ws   = (float*)d_ws;
    float* XP   = ws;                                  // MROWS*NCOL floats (8 MB)
    float* Wall = XP + (size_t)MROWS * NCOL;           // 32*64
    float* Whid = Wall + NCOL * INPUT_DIM;             // 32*8
    float* bsum = Whid + NCOL * NQ;                    // 32

    qlstm_pack<<<1, 256, 0, stream>>>(Wf, bf, thf, Wi, bi, thi, Wu, bu, thu,
                                      Wo, bo, tho, Wall, Whid, bsum);

#ifdef HAVE_WMMA_F32X4
    // 4096 waves, 8 waves (256 threads) per block
    qlstm_xproj_wmma<<<(MROWS / 16) / 8, 256, 0, stream>>>(X, Wall, bsum, XP);
#else
    qlstm_xproj_plain<<<(MROWS * NCOL) / 256, 256, 0, stream>>>(X, Wall, bsum, XP);
#endif

    qlstm_recurrence<<<BATCH / 256, 256, 0, stream>>>(XP, Whid, (float*)d_out);
}